// PooledTransformerDecoder_3839700763251
// MI455X (gfx1250) — compile-verified
//
#include <hip/hip_runtime.h>
#include <math.h>

#define B_    4
#define E_    512
#define H_    8
#define HD_   64
#define HID_  2048
#define NL_   2
#define V_    32000
#define L_    32
#define SCALE_ 0.125f   /* 1/sqrt(64) */

typedef __attribute__((ext_vector_type(2))) float v2f;
typedef __attribute__((ext_vector_type(8))) float v8f;

// ---------------------------------------------------------------------------
// Generic GEMM: Y[M,N] = X[M,K] @ W[N,K]^T + bias  (optional relu)
// One wave computes one 16x16 output tile via V_WMMA_F32_16X16X4_F32.
// 4 waves per block sweep consecutive N tiles. N must be a multiple of 16
// (true for all uses here: 512 / 1536 / 2048 / 32000).
// Out-of-range A rows are CLAMPED (not masked): row m of D depends only on
// row m of A, and stores are guarded by m < M, so garbage rows are harmless.
// This keeps EXEC all-1s and branch-free through the K loop.
// Two independent accumulators break the WMMA RAW chain.
// ---------------------------------------------------------------------------
__global__ __launch_bounds__(128) void wmma_gemm_nt(
    const float* __restrict__ X, const float* __restrict__ W,
    const float* __restrict__ bias, float* __restrict__ Y,
    int M, int N, int K, int relu) {
  const int lane  = threadIdx.x & 31;
  const int ntile = blockIdx.x * 4 + (threadIdx.x >> 5);
  if (ntile * 16 >= N) return;            // uniform per wave -> EXEC stays all-1s
  const int col = lane & 15;
  const int kh  = (lane >> 4) * 2;        // lanes 0-15: K pair 0,1 ; lanes 16-31: K pair 2,3

  int arow = blockIdx.y * 16 + col;
  if (arow >= M) arow = M - 1;            // clamp: garbage rows never stored
  const float* xrow = X + (size_t)arow * K + kh;
  const float* wrow = W + (size_t)(ntile * 16 + col) * K + kh;  // B[k][n] = W[n][k]

  v8f acc0 = {}, acc1 = {};
  #pragma unroll 4
  for (int k = 0; k < K; k += 8) {
    v2f a0 = *(const v2f*)(xrow + k);
    v2f b0 = *(const v2f*)(wrow + k);
    v2f a1 = *(const v2f*)(xrow + k + 4);
    v2f b1 = *(const v2f*)(wrow + k + 4);
    // (neg_a, A, neg_b, B, c_mod, C, reuse_a, reuse_b)
    acc0 = __builtin_amdgcn_wmma_f32_16x16x4_f32(false, a0, false, b0,
                                                 (short)0, acc0, false, false);
    acc1 = __builtin_amdgcn_wmma_f32_16x16x4_f32(false, a1, false, b1,
                                                 (short)0, acc1, false, false);
  }
  v8f acc = acc0 + acc1;

  const int   nIdx  = ntile * 16 + col;
  const float bv    = bias ? bias[nIdx] : 0.0f;
  const int   mbase = blockIdx.y * 16 + (lane >> 4) * 8; // C/D: VGPR r holds M = r + 8*(lane>=16)
  #pragma unroll
  for (int r = 0; r < 8; ++r) {
    int m = mbase + r;
    if (m < M) {
      float v = acc[r] + bv;
      if (relu) v = fmaxf(v, 0.0f);
      Y[(size_t)m * N + nIdx] = v;
    }
  }
}

// ---------------------------------------------------------------------------
// tokens[0][b] = 0  (sos)
// ---------------------------------------------------------------------------
__global__ void init_tokens_kernel(int* tokens) {
  if (threadIdx.x < B_) tokens[threadIdx.x] = 0;
}

// ---------------------------------------------------------------------------
// X[t,b,:] = emb[tokens[t,b]] + pe[t]   for rows t*B+b < n*B
// Sinusoidal PE computed in-kernel.
// ---------------------------------------------------------------------------
__global__ __launch_bounds__(128) void embed_kernel(
    const int* __restrict__ tokens, const float* __restrict__ emb,
    float* __restrict__ X) {
  const int row = blockIdx.x;          // row = t*B + b
  const int t = row >> 2, b = row & 3;
  const int tok = tokens[t * B_ + b];
  const float c = -logf(10000.0f) / (float)E_;
  for (int i = threadIdx.x; i < E_; i += blockDim.x) {
    float div = expf((float)((i >> 1) * 2) * c);
    float ang = (float)t * div;
    float pe  = (i & 1) ? cosf(ang) : sinf(ang);
    X[(size_t)row * E_ + i] = emb[(size_t)tok * E_ + i] + pe;
  }
}

// ---------------------------------------------------------------------------
// Self attention for one (b,h): scores = Q K^T * scale, softmax over all
// n (=s+1) keys (mask already encoded by truncating to n), O = A V.
// Thread tq handles query row tq.  QKV layout: [n*B, 3E], q|k|v slices.
// ---------------------------------------------------------------------------
__global__ __launch_bounds__(64) void attn_kernel(
    const float* __restrict__ QKV, float* __restrict__ O, int n) {
  const int b = blockIdx.x & 3;
  const int h = blockIdx.x >> 2;
  const int tq = threadIdx.x;
  if (tq >= n) return;

  const float* qp = QKV + ((size_t)(tq * B_ + b)) * (3 * E_) + h * HD_;
  float sc[L_ + 1];
  float mx = -1e30f;
  for (int tk = 0; tk < n; ++tk) {
    const float* kp = QKV + ((size_t)(tk * B_ + b)) * (3 * E_) + E_ + h * HD_;
    float d = 0.0f;
    #pragma unroll
    for (int i = 0; i < HD_; ++i) d += qp[i] * kp[i];
    d *= SCALE_;
    sc[tk] = d;
    mx = fmaxf(mx, d);
  }
  float sum = 0.0f;
  for (int tk = 0; tk < n; ++tk) { sc[tk] = expf(sc[tk] - mx); sum += sc[tk]; }
  const float inv = 1.0f / sum;

  float o[HD_];
  #pragma unroll
  for (int i = 0; i < HD_; ++i) o[i] = 0.0f;
  for (int tk = 0; tk < n; ++tk) {
    const float a = sc[tk] * inv;
    const float* vp = QKV + ((size_t)(tk * B_ + b)) * (3 * E_) + 2 * E_ + h * HD_;
    #pragma unroll
    for (int i = 0; i < HD_; ++i) o[i] += a * vp[i];
  }
  float* op = O + ((size_t)(tq * B_ + b)) * E_ + h * HD_;
  #pragma unroll
  for (int i = 0; i < HD_; ++i) op[i] = o[i];
}

// ---------------------------------------------------------------------------
// block reduction (blockDim.x == 128)
// ---------------------------------------------------------------------------
__device__ __forceinline__ float block_reduce_sum(float v, float* sbuf) {
  const int tid = threadIdx.x;
  sbuf[tid] = v;
  __syncthreads();
  for (int off = 64; off > 0; off >>= 1) {
    if (tid < off) sbuf[tid] += sbuf[tid + off];
    __syncthreads();
  }
  float r = sbuf[0];
  __syncthreads();
  return r;
}

// ---------------------------------------------------------------------------
// Out = LN2( LN1(X + Yp) + ca[b] )      one block (128 thr) per row
// ---------------------------------------------------------------------------
__global__ __launch_bounds__(128) void res_ln_ln_kernel(
    const float* __restrict__ X, const float* __restrict__ Yp,
    const float* __restrict__ g1, const float* __restrict__ b1,
    const float* __restrict__ ca, const float* __restrict__ g2,
    const float* __restrict__ b2, float* __restrict__ Out) {
  __shared__ float sbuf[128];
  const int row = blockIdx.x;
  const int b = row & 3;
  const int tid = threadIdx.x;

  float v[4];
  float s = 0.0f;
  #pragma unroll
  for (int i = 0; i < 4; ++i) {
    int idx = tid + i * 128;
    v[i] = X[(size_t)row * E_ + idx] + Yp[(size_t)row * E_ + idx];
    s += v[i];
  }
  s = block_reduce_sum(s, sbuf);
  const float m1 = s * (1.0f / E_);
  float q = 0.0f;
  #pragma unroll
  for (int i = 0; i < 4; ++i) { float d = v[i] - m1; q += d * d; }
  q = block_reduce_sum(q, sbuf);
  const float inv1 = rsqrtf(q * (1.0f / E_) + 1e-5f);

  float w[4];
  float s2 = 0.0f;
  #pragma unroll
  for (int i = 0; i < 4; ++i) {
    int idx = tid + i * 128;
    float u = (v[i] - m1) * inv1 * g1[idx] + b1[idx];
    w[i] = u + ca[(size_t)b * E_ + idx];
    s2 += w[i];
  }
  s2 = block_reduce_sum(s2, sbuf);
  const float m2 = s2 * (1.0f / E_);
  float q2 = 0.0f;
  #pragma unroll
  for (int i = 0; i < 4; ++i) { float d = w[i] - m2; q2 += d * d; }
  q2 = block_reduce_sum(q2, sbuf);
  const float inv2 = rsqrtf(q2 * (1.0f / E_) + 1e-5f);
  #pragma unroll
  for (int i = 0; i < 4; ++i) {
    int idx = tid + i * 128;
    Out[(size_t)row * E_ + idx] = (w[i] - m2) * inv2 * g2[idx] + b2[idx];
  }
}

// ---------------------------------------------------------------------------
// Out = LN3(X + F)                      one block (128 thr) per row
// ---------------------------------------------------------------------------
__global__ __launch_bounds__(128) void res_ln_kernel(
    const float* __restrict__ X, const float* __restrict__ F,
    const float* __restrict__ g, const float* __restrict__ b,
    float* __restrict__ Out) {
  __shared__ float sbuf[128];
  const int row = blockIdx.x;
  const int tid = threadIdx.x;
  float v[4];
  float s = 0.0f;
  #pragma unroll
  for (int i = 0; i < 4; ++i) {
    int idx = tid + i * 128;
    v[i] = X[(size_t)row * E_ + idx] + F[(size_t)row * E_ + idx];
    s += v[i];
  }
  s = block_reduce_sum(s, sbuf);
  const float m = s * (1.0f / E_);
  float q = 0.0f;
  #pragma unroll
  for (int i = 0; i < 4; ++i) { float d = v[i] - m; q += d * d; }
  q = block_reduce_sum(q, sbuf);
  const float inv = rsqrtf(q * (1.0f / E_) + 1e-5f);
  #pragma unroll
  for (int i = 0; i < 4; ++i) {
    int idx = tid + i * 128;
    Out[(size_t)row * E_ + idx] = (v[i] - m) * inv * g[idx] + b[idx];
  }
}

// ---------------------------------------------------------------------------
// argmax over V per batch row; first-max-index tie break (matches jnp.argmax).
// Writes next token id (int, ws) and float token into d_out token area.
// ---------------------------------------------------------------------------
__global__ __launch_bounds__(256) void argmax_kernel(
    const float* __restrict__ logits, int* __restrict__ tokens,
    float* __restrict__ tok_out, int s) {
  __shared__ float sval[256];
  __shared__ int   sidx[256];
  const int b = blockIdx.x;
  const int tid = threadIdx.x;
  const float* lp = logits + (size_t)b * V_;
  float best = -3.4e38f;
  int bi = V_;
  for (int i = tid; i < V_; i += 256) {
    float v = lp[i];
    if (v > best) { best = v; bi = i; }   // per-thread indices are increasing
  }
  sval[tid] = best; sidx[tid] = bi;
  __syncthreads();
  for (int off = 128; off > 0; off >>= 1) {
    if (tid < off) {
      float v2 = sval[tid + off]; int i2 = sidx[tid + off];
      if (v2 > sval[tid] || (v2 == sval[tid] && i2 < sidx[tid])) {
        sval[tid] = v2; sidx[tid] = i2;
      }
    }
    __syncthreads();
  }
  if (tid == 0) {
    tokens[(s + 1) * B_ + b] = sidx[0];
    tok_out[b * L_ + s] = (float)sidx[0];   // tokens[1:].T -> [B,L]
  }
}

// ---------------------------------------------------------------------------
extern "C" void kernel_launch(void* const* d_in, const int* in_sizes, int n_in,
                              void* d_out, int out_size, void* d_ws, size_t ws_size,
                              hipStream_t stream) {
  const float* memory   = (const float*)d_in[0];    // [B,E]
  const float* emb      = (const float*)d_in[1];    // [V,E]
  const float* sa_in_w  = (const float*)d_in[2];    // [NL,3E,E]
  const float* sa_in_b  = (const float*)d_in[3];    // [NL,3E]
  const float* sa_out_w = (const float*)d_in[4];    // [NL,E,E]
  const float* sa_out_b = (const float*)d_in[5];    // [NL,E]
  const float* ca_in_w  = (const float*)d_in[6];
  const float* ca_in_b  = (const float*)d_in[7];
  const float* ca_out_w = (const float*)d_in[8];
  const float* ca_out_b = (const float*)d_in[9];
  const float* ln1_w = (const float*)d_in[10];
  const float* ln1_b = (const float*)d_in[11];
  const float* ln2_w = (const float*)d_in[12];
  const float* ln2_b = (const float*)d_in[13];
  const float* ln3_w = (const float*)d_in[14];
  const float* ln3_b = (const float*)d_in[15];
  const float* ff1_w = (const float*)d_in[16];      // [NL,HID,E]
  const float* ff1_b = (const float*)d_in[17];
  const float* ff2_w = (const float*)d_in[18];      // [NL,E,HID]
  const float* ff2_b = (const float*)d_in[19];
  const float* out_w = (const float*)d_in[20];      // [V,E]
  const float* out_b = (const float*)d_in[21];

  float* outp    = (float*)d_out;                   // [0,128) tokens, [128,...) logits
  float* logits0 = outp + B_ * L_;

  // workspace layout
  int*   tokens = (int*)d_ws;                       // (L+1)*B ints
  float* f      = (float*)d_ws;
  float* Xbuf   = f + 256;
  float* X2     = Xbuf + (L_ + 1) * B_ * E_;        // 67584
  float* QKV    = X2   + (L_ + 1) * B_ * E_;
  float* Obuf   = QKV  + (L_ + 1) * B_ * 3 * E_;    // 202752
  float* Hf     = Obuf + (L_ + 1) * B_ * E_;
  float* F2     = Hf   + (L_ + 1) * B_ * HID_;      // 270336
  float* catmp  = F2   + (L_ + 1) * B_ * E_;
  float* cac    = catmp + B_ * E_;                  // ca_const [NL][B][E]

  hipLaunchKernelGGL(init_tokens_kernel, dim3(1), dim3(32), 0, stream, tokens);

  // Precompute cross-attention constants per layer:
  //   ca_const = (mem @ Wv^T + bv) @ Wout^T + bout   (softmax over 1 key == 1)
  for (int l = 0; l < NL_; ++l) {
    const float* wv = ca_in_w + (size_t)l * 3 * E_ * E_ + (size_t)2 * E_ * E_;
    const float* bv = ca_in_b + (size_t)l * 3 * E_ + 2 * E_;
    hipLaunchKernelGGL(wmma_gemm_nt, dim3(E_ / 64, 1), dim3(128), 0, stream,
                       memory, wv, bv, catmp, B_, E_, E_, 0);
    hipLaunchKernelGGL(wmma_gemm_nt, dim3(E_ / 64, 1), dim3(128), 0, stream,
                       catmp, ca_out_w + (size_t)l * E_ * E_,
                       ca_out_b + (size_t)l * E_, cac + (size_t)l * B_ * E_,
                       B_, E_, E_, 0);
  }

  for (int s = 0; s < L_; ++s) {
    const int n = s + 1;
    const int M = n * B_;
    const int mt = (M + 15) / 16;

    hipLaunchKernelGGL(embed_kernel, dim3(M), dim3(128), 0, stream,
                       tokens, emb, Xbuf);

    for (int l = 0; l < NL_; ++l) {
      // QKV projection: [M,E] @ [3E,E]^T
      hipLaunchKernelGGL(wmma_gemm_nt, dim3(3 * E_ / 64, mt), dim3(128), 0, stream,
                         Xbuf, sa_in_w + (size_t)l * 3 * E_ * E_,
                         sa_in_b + (size_t)l * 3 * E_, QKV, M, 3 * E_, E_, 0);
      // attention per (b,h)
      hipLaunchKernelGGL(attn_kernel, dim3(B_ * H_), dim3(64), 0, stream,
                         QKV, Obuf, n);
      // out projection -> F2 (reuse as scratch)
      hipLaunchKernelGGL(wmma_gemm_nt, dim3(E_ / 64, mt), dim3(128), 0, stream,
                         Obuf, sa_out_w + (size_t)l * E_ * E_,
                         sa_out_b + (size_t)l * E_, F2, M, E_, E_, 0);
      // x2 = LN2( LN1(x + attn) + ca_const )
      hipLaunchKernelGGL(res_ln_ln_kernel, dim3(M), dim3(128), 0, stream,
                         Xbuf, F2, ln1_w + (size_t)l * E_, ln1_b + (size_t)l * E_,
                         cac + (size_t)l * B_ * E_,
                         ln2_w + (size_t)l * E_, ln2_b + (size_t)l * E_, X2);
      // FFN
      hipLaunchKernelGGL(wmma_gemm_nt, dim3(HID_ / 64, mt), dim3(128), 0, stream,
                         X2, ff1_w + (size_t)l * HID_ * E_,
                         ff1_b + (size_t)l * HID_, Hf, M, HID_, E_, 1);
      hipLaunchKernelGGL(wmma_gemm_nt, dim3(E_ / 64, mt), dim3(128), 0, stream,
                         Hf, ff2_w + (size_t)l * E_ * HID_,
                         ff2_b + (size_t)l * E_, F2, M, E_, HID_, 0);
      // x = LN3(x2 + ffn)
      hipLaunchKernelGGL(res_ln_kernel, dim3(M), dim3(128), 0, stream,
                         X2, F2, ln3_w + (size_t)l * E_, ln3_b + (size_t)l * E_, Xbuf);
    }

    // logits for position s: rows [s*B, s*B+4) of Xbuf
    float* lg = logits0 + (size_t)s * B_ * V_;
    hipLaunchKernelGGL(wmma_gemm_nt, dim3(V_ / 64, 1), dim3(128), 0, stream,
                       Xbuf + (size_t)s * B_ * E_, out_w, out_b, lg, B_, V_, E_, 0);
    hipLaunchKernelGGL(argmax_kernel, dim3(B_), dim3(256), 0, stream,
                       lg, tokens, outp, s);
  }
}